// Fp8Unpadding_45217415692550
// MI455X (gfx1250) — compile-verified
//
#include <hip/hip_runtime.h>

// Pure streaming row-gather: out[j, :] = inp[gather_idx[j], :].
// Memory-bound (0 FLOPs): 2.16 GB total traffic -> ~93 us floor at 23.3 TB/s.
// Strategy: 1 workgroup per output row, b128 non-temporal loads/stores
// (dataset >> 192 MB L2, zero reuse), 4 outstanding b128 loads per thread.

typedef __attribute__((ext_vector_type(4))) float v4f;

__global__ __launch_bounds__(256) void gather_rows_nt_kernel(
    const float* __restrict__ inp,
    const int*   __restrict__ gather_idx,
    float*       __restrict__ out,
    int hidden4)  // hidden / 4 (number of float4 per row)
{
    const int row = blockIdx.x;                 // output row (uniform)
    const long long src = gather_idx[row];      // uniform -> scalar load

    const v4f* __restrict__ s = (const v4f*)inp + src * (long long)hidden4;
    v4f*       __restrict__ d = (v4f*)out + (long long)row * (long long)hidden4;

    const int stride = 256;                     // blockDim.x (launch_bounds)
    int c = threadIdx.x;

    // Main body: 4 b128 loads in flight before the stores (deep LOADcnt).
    // For hidden=4096 this runs exactly once (1024 float4 per row).
    for (; c + 3 * stride < hidden4; c += 4 * stride) {
        v4f v0 = __builtin_nontemporal_load(s + c);
        v4f v1 = __builtin_nontemporal_load(s + c + stride);
        v4f v2 = __builtin_nontemporal_load(s + c + 2 * stride);
        v4f v3 = __builtin_nontemporal_load(s + c + 3 * stride);
        __builtin_nontemporal_store(v0, d + c);
        __builtin_nontemporal_store(v1, d + c + stride);
        __builtin_nontemporal_store(v2, d + c + 2 * stride);
        __builtin_nontemporal_store(v3, d + c + 3 * stride);
    }
    // Remainder (not taken for hidden=4096, kept for generality).
    for (; c < hidden4; c += stride) {
        v4f v = __builtin_nontemporal_load(s + c);
        __builtin_nontemporal_store(v, d + c);
    }
}

extern "C" void kernel_launch(void* const* d_in, const int* in_sizes, int n_in,
                              void* d_out, int out_size, void* d_ws, size_t ws_size,
                              hipStream_t stream) {
    (void)n_in; (void)d_ws; (void)ws_size;

    const float* inp        = (const float*)d_in[0];  // [total_padded, hidden] fp32
    const int*   gather_idx = (const int*)d_in[1];    // [rows] int32
    float*       out        = (float*)d_out;          // [rows, hidden] fp32

    const int rows    = in_sizes[1];          // number of gathered output rows
    const int hidden  = out_size / rows;      // 4096 in the reference
    const int hidden4 = hidden >> 2;          // float4 per row (1024)

    if (rows <= 0 || hidden4 <= 0) return;

    dim3 grid((unsigned)rows);
    dim3 block(256);
    gather_rows_nt_kernel<<<grid, block, 0, stream>>>(inp, gather_idx, out, hidden4);
}